// DCT2net_60662118089392
// MI455X (gfx1250) — compile-verified
//
#include <hip/hip_runtime.h>

typedef float v2f __attribute__((ext_vector_type(2)));
typedef float v8f __attribute__((ext_vector_type(8)));

#define P      13
#define PP     169           // 13*13
#define PPAD   176           // 11*16
#define IMG    512
#define GRIDP  500           // 512 - 13 + 1 patch grid
#define OUTD   488           // 500 - 12
#define TY     16            // tile rows of patch grid per block
#define TX     8             // tile cols of patch grid per block
#define XH     (TY + P - 1)  // 28
#define XW     (TX + P - 1)  // 20
#define NCHUNK 43            // ceil(169/4) K-chunks of 4 for fp32 WMMA
#define KPAD   (NCHUNK * 4)  // 172

// workspace layout (in floats)
#define WF_OFF   0
#define WI_OFF   (PPAD * PPAD)                 // 30976
#define ACC_OFF  (2 * PPAD * PPAD)             // 61952
#define WIMG_OFF (ACC_OFF + 2 * IMG * IMG)     // 61952 + 524288

// ---------------------------------------------------------------------------
// Kernel 0: build padded forward/inverse DCT matrices, zero the accumulator.
// The 169x169 DCT basis is orthonormal -> inverse == transpose.
// ---------------------------------------------------------------------------
__global__ __launch_bounds__(256) void dct_prep(const float* __restrict__ Wdct,
                                                float* __restrict__ ws) {
  int idx = blockIdx.x * 256 + threadIdx.x;
  if (idx < PPAD * PPAD) {
    int r = idx / PPAD, c = idx - r * PPAD;
    bool in = (r < PP) && (c < PP);
    ws[WF_OFF + idx] = in ? Wdct[r * PP + c] : 0.0f;  // forward:  T = Wf * patch
    ws[WI_OFF + idx] = in ? Wdct[c * PP + r] : 0.0f;  // inverse:  Z = Wf^T * y
  }
  if (idx < 2 * IMG * IMG) ws[ACC_OFF + idx] = 0.0f;
}

// ---------------------------------------------------------------------------
// Kernel 1: fused DCT -> shrinkage -> inverse DCT -> overlap-add.
// One block = 8 waves = 16x8 tile of patch positions (128 GEMM columns).
// Wave w owns columns [16w, 16w+16); it holds every K-row of its columns,
// so the D-layout -> B-layout handoff between the two GEMMs is wave-private.
// The fold (overlap-add) is staged in LDS with ds_add_f32 and flushed with
// one global atomic per tile pixel (38x fewer global atomics).
// ---------------------------------------------------------------------------
__global__ __launch_bounds__(256) void dct_main(const float* __restrict__ x,
                                                const float* __restrict__ sigma,
                                                float* __restrict__ ws) {
  __shared__ float xs[XH * XW];                       // input patch tile
  __shared__ __align__(16) float ys[8][16][16];       // per-wave Y panel [n][k]
  __shared__ float zacc[XH * XW];                     // LDS overlap-add tile
  __shared__ int   ktab[KPAD];                        // k -> i*XW + j offsets

  const float* __restrict__ Wf = ws + WF_OFF;
  const float* __restrict__ Wi = ws + WI_OFF;
  float* __restrict__ acc  = ws + ACC_OFF;
  float* __restrict__ wimg = ws + WIMG_OFF;

  const int tid  = threadIdx.x;
  const int wave = tid >> 5;
  const int lane = tid & 31;
  const int l15  = lane & 15;
  const int hi2  = (lane & 16) ? 2 : 0;  // K offset inside A/B fragments
  const int hi8  = (lane & 16) ? 8 : 0;  // M offset inside C/D fragments

  const int b  = blockIdx.z;
  const int h0 = blockIdx.y * TY;
  const int w0 = blockIdx.x * TX;

  // stage the (TY+12) x (TX+12) input region into LDS (zero-fill OOB),
  // zero the fold accumulator, and build the k -> patch-offset table
  for (int idx = tid; idx < XH * XW; idx += 256) {
    int r = idx / XW, c = idx - r * XW;
    int gh = h0 + r, gw = w0 + c;
    xs[idx]   = (gh < IMG && gw < IMG) ? x[(b * IMG + gh) * IMG + gw] : 0.0f;
    zacc[idx] = 0.0f;
  }
  if (tid < KPAD) {
    int q = tid / 13;
    ktab[tid] = q * XW + (tid - 13 * q);
  }
  __syncthreads();

  const float inv3s = 1.0f / (3.0f * sigma[0]);

  const int n  = wave * 16 + l15;  // GEMM column = pixel within tile
  const int py = n >> 3;
  const int px = n & 7;
  const int h  = h0 + py;
  const int w_ = w0 + px;
  const int pbase = py * XW + px;
  const bool valid = (h < GRIDP) && (w_ < GRIDP);

  // ----------------- forward GEMM: T(176 x 128) = Wf * patches -------------
  v8f Cf[11] = {};
  for (int kc = 0; kc < NCHUNK; ++kc) {
    const int kb = kc * 4 + hi2;
    v2f bfr;
    {
      int o0 = ktab[kb];
      int o1 = ktab[kb + 1];
      bfr.x = (kb < PP)     ? xs[pbase + o0] : 0.0f;
      bfr.y = (kb + 1 < PP) ? xs[pbase + o1] : 0.0f;
    }
#pragma unroll
    for (int m = 0; m < 11; ++m) {
      v2f afr = *(const v2f*)(Wf + (m * 16 + l15) * PPAD + kb);
      Cf[m] = __builtin_amdgcn_wmma_f32_16x16x4_f32(false, afr, false, bfr,
                                                    (short)0, Cf[m], false, false);
    }
  }

  // ----------------- shrinkage: nz = udiff(t/(3*sigma)), y = t*nz ----------
  float nzsum = 0.0f;
#pragma unroll
  for (int m = 0; m < 11; ++m) {
#pragma unroll
    for (int r = 0; r < 8; ++r) {
      float t  = Cf[m][r];
      float u  = t * inv3s;
      float au = fabsf(u);
      float us = (au < 1.3f) ? u : 0.0f;
      float p2 = us * us;          // ^2
      p2 = p2 * p2;                // ^4
      p2 = p2 * p2;                // ^8
      p2 = p2 * p2;                // ^16
      p2 = p2 * p2;                // ^32
      p2 = p2 * p2;                // ^64
      float nz = (au >= 1.3f) ? 1.0f : p2 / (p2 + 1.0f);
      nzsum += nz;
      Cf[m][r] = t * nz;           // y = t * nz (weight applied later)
    }
  }
  // rows of a column are split between lane n and lane n+16 -> one xor-shuffle
  nzsum += __shfl_xor(nzsum, 16, 32);
  const float wpix = 1.0f / (1.0f + nzsum);

  if (lane < 16 && valid) wimg[(b * GRIDP + h) * GRIDP + w_] = wpix;

  // ----------------- inverse GEMM: Z = Wi * (wpix * Y) ---------------------
  v8f Cz[11] = {};
#pragma unroll
  for (int mp = 0; mp < 11; ++mp) {
    // stage this wave's 16x16 Y panel (D-layout -> [n][k] LDS layout)
#pragma unroll
    for (int r = 0; r < 8; ++r) {
      ys[wave][l15][r + hi8] = Cf[mp][r] * wpix;
    }
    __syncthreads();
#pragma unroll
    for (int c = 0; c < 4; ++c) {
      const int kloc = c * 4 + hi2;
      v2f bfr = *(const v2f*)&ys[wave][l15][kloc];
      const int ka = mp * 16 + c * 4 + hi2;
#pragma unroll
      for (int mz = 0; mz < 11; ++mz) {
        v2f afr = *(const v2f*)(Wi + (mz * 16 + l15) * PPAD + ka);
        Cz[mz] = __builtin_amdgcn_wmma_f32_16x16x4_f32(false, afr, false, bfr,
                                                       (short)0, Cz[mz], false, false);
      }
    }
    __syncthreads();
  }

  // ----------------- overlap-add: LDS fold, then one global atomic/pixel ---
  if (valid) {
#pragma unroll
    for (int mz = 0; mz < 11; ++mz) {
#pragma unroll
      for (int r = 0; r < 8; ++r) {
        int cidx = mz * 16 + r + hi8;
        if (cidx < PP) {
          atomicAdd(&zacc[pbase + ktab[cidx]], Cz[mz][r]);   // ds_add_f32
        }
      }
    }
  }
  __syncthreads();

  for (int idx = tid; idx < XH * XW; idx += 256) {
    int r = idx / XW, c = idx - r * XW;
    int gh = h0 + r, gw = w0 + c;
    if (gh < IMG && gw < IMG) {
      atomicAdd(&acc[(b * IMG + gh) * IMG + gw], zacc[idx]);
    }
  }
}

// ---------------------------------------------------------------------------
// Kernel 2: out = acc / (13x13 box-sum of w)
// ---------------------------------------------------------------------------
__global__ __launch_bounds__(256) void dct_finalize(const float* __restrict__ ws_ro,
                                                    float* __restrict__ out) {
  const float* __restrict__ acc  = ws_ro + ACC_OFF;
  const float* __restrict__ wimg = ws_ro + WIMG_OFF;

  int idx = blockIdx.x * 256 + threadIdx.x;
  if (idx >= 2 * OUTD * OUTD) return;
  int b  = idx / (OUTD * OUTD);
  int rr = idx - b * (OUTD * OUTD);
  int o1 = rr / OUTD;
  int o2 = rr - o1 * OUTD;

  float s = 0.0f;
#pragma unroll
  for (int u = 0; u < P; ++u) {
    const float* rowp = wimg + ((b * GRIDP + o1 + u) * GRIDP + o2);
#pragma unroll
    for (int v = 0; v < P; ++v) s += rowp[v];
  }
  float num = acc[(b * IMG + o1 + P - 1) * IMG + (o2 + P - 1)];
  out[idx] = num / s;
}

// ---------------------------------------------------------------------------
extern "C" void kernel_launch(void* const* d_in, const int* in_sizes, int n_in,
                              void* d_out, int out_size, void* d_ws, size_t ws_size,
                              hipStream_t stream) {
  const float* x     = (const float*)d_in[0];  // (2,1,512,512) fp32
  const float* sigma = (const float*)d_in[1];  // (1,) fp32
  const float* Wdct  = (const float*)d_in[2];  // (169,1,13,13) fp32
  float* ws = (float*)d_ws;

  // Kernel 0: weights + accumulator init  (covers max(176*176, 2*512*512))
  {
    int nelem = 2 * IMG * IMG;
    dct_prep<<<(nelem + 255) / 256, 256, 0, stream>>>(Wdct, ws);
  }

  // Kernel 1: fused WMMA pipeline over the 500x500 patch grid, 2 batches
  {
    dim3 grid((GRIDP + TX - 1) / TX, (GRIDP + TY - 1) / TY, 2);  // 63 x 32 x 2
    dct_main<<<grid, 256, 0, stream>>>(x, sigma, ws);
  }

  // Kernel 2: normalize into d_out (2*488*488 fp32)
  {
    int nelem = 2 * OUTD * OUTD;
    dct_finalize<<<(nelem + 255) / 256, 256, 0, stream>>>(ws, (float*)d_out);
  }
}